// GATLayerWithEdgeFeatures_17093969838499
// MI455X (gfx1250) — compile-verified
//
#include <hip/hip_runtime.h>
#include <math.h>

// Problem constants (match reference): N=50000, IN_F=128, E=800000, EDGE_F=32, H=4, C=32
#define IN_F 128
#define HC 128
#define HEADS 4
#define CH 32
#define EDGE_F_ 32
#define NEG_SLOPE 0.2f
#define EPS_ 1e-16f

typedef float v2f __attribute__((ext_vector_type(2)));
typedef float v8f __attribute__((ext_vector_type(8)));

// ---------------------------------------------------------------------------
// Kernel 1: fold att_edge into W_edge:  v[h][f] = sum_c att_edge[h,c]*W_edge[h*32+c, f]
// ---------------------------------------------------------------------------
__global__ void fold_edge_kernel(const float* __restrict__ We,
                                 const float* __restrict__ att_edge,
                                 float* __restrict__ v_edge) {
    int t = threadIdx.x;            // 128 threads
    int h = t >> 5, f = t & 31;
    float s = 0.f;
    #pragma unroll
    for (int c = 0; c < CH; ++c)
        s += att_edge[h * CH + c] * We[(size_t)(h * CH + c) * EDGE_F_ + f];
    v_edge[h * EDGE_F_ + f] = s;
}

// ---------------------------------------------------------------------------
// Kernel 2: init segment-max (monotonic uint encoding of -inf) and denom=0
// encode(f): u=bits(f); sign? ~u : u|0x80000000  (order preserving)
// ---------------------------------------------------------------------------
__global__ void init_kernel(unsigned* __restrict__ nmax, float* __restrict__ denom, int n) {
    int i = blockIdx.x * blockDim.x + threadIdx.x;
    if (i < n) { nmax[i] = 0x007FFFFFu; /* encode(-inf) */ denom[i] = 0.f; }
}

// ---------------------------------------------------------------------------
// Kernel 3: node GEMMs via V_WMMA_F32_16X16X4_F32 (full fp32 precision).
//   x    = nf @ W_gat.T            -> ws (and LDS for the dot pass)
//   out  = nf @ W_skip.T + bias    -> d_out (scatter accumulator base)
//   a_src[n,h] = x[n,h,:]·att_src[h,:],  a_dst likewise.
// One block = 128 threads (4 waves) handles a 16-row tile. N % 16 == 0.
// ---------------------------------------------------------------------------
__global__ void __launch_bounds__(128)
node_gemm_kernel(const float* __restrict__ nf,
                 const float* __restrict__ Wg,
                 const float* __restrict__ Ws,
                 const float* __restrict__ att_src,
                 const float* __restrict__ att_dst,
                 const float* __restrict__ bias,
                 float* __restrict__ x,
                 float* __restrict__ a_src,
                 float* __restrict__ a_dst,
                 float* __restrict__ outv) {
    __shared__ float sA[16][IN_F];      // node-feature tile
    __shared__ float sX[16][HC + 4];    // x tile (padded), for the a_src/a_dst dots

    const int rowBase = blockIdx.x * 16;
    const int tid  = threadIdx.x;
    const int wave = tid >> 5;
    const int lane = tid & 31;

    // cooperative float4 load of the 16x128 A tile
    for (int i = tid; i < 16 * (IN_F / 4); i += 128) {
        int r = i >> 5;          // row 0..15
        int c4 = i & 31;         // float4 index 0..31
        float4 v = ((const float4*)(nf + (size_t)(rowBase + r) * IN_F))[c4];
        ((float4*)&sA[r][0])[c4] = v;
    }
    __syncthreads();

    // WMMA fragment addressing (ISA 7.12.2, 32-bit 16x4 A / 4x16 B / 16x16 D):
    const int mrow = lane & 15;          // A: M index
    const int koff = (lane >> 4) * 2;    // A/B: K sub-block {0,1} vs {2,3}
    const int ncol = lane & 15;          // B/D: N index
    const int rbase = (lane >> 4) * 8;   // D: M base (0 or 8)

    for (int m = 0; m < 2; ++m) {            // 0: W_gat -> x ; 1: W_skip -> out
        const float* W = m ? Ws : Wg;
        for (int t = 0; t < 2; ++t) {
            const int jt = wave * 2 + t;     // J tile 0..7
            const int j  = jt * 16 + ncol;   // this lane's output column
            v8f acc = {};
            #pragma unroll 8
            for (int k = 0; k < IN_F; k += 4) {
                v2f a, b;
                const float2 av = *(const float2*)&sA[mrow][k + koff];
                const float2 bv = *(const float2*)(W + (size_t)j * IN_F + k + koff); // B[k',j]=W[j,k']
                a[0] = av.x; a[1] = av.y;
                b[0] = bv.x; b[1] = bv.y;
                acc = __builtin_amdgcn_wmma_f32_16x16x4_f32(
                          false, a, false, b, (short)0, acc, false, false);
            }
            if (m == 0) {
                #pragma unroll
                for (int r = 0; r < 8; ++r) {
                    const int row = rbase + r, col = jt * 16 + ncol;
                    x[(size_t)(rowBase + row) * HC + col] = acc[r];
                    sX[row][col] = acc[r];
                }
            } else {
                #pragma unroll
                for (int r = 0; r < 8; ++r) {
                    const int row = rbase + r, col = jt * 16 + ncol;
                    outv[(size_t)(rowBase + row) * HC + col] = acc[r] + bias[col];
                }
            }
        }
    }
    __syncthreads();

    // per-head attention dots for this row tile: 16 rows x 4 heads
    if (tid < 64) {
        const int r = tid >> 2, h = tid & 3;
        float ss = 0.f, sd = 0.f;
        #pragma unroll
        for (int c = 0; c < CH; ++c) {
            const float xv = sX[r][h * CH + c];
            ss += xv * att_src[h * CH + c];
            sd += xv * att_dst[h * CH + c];
        }
        a_src[(size_t)(rowBase + r) * HEADS + h] = ss;
        a_dst[(size_t)(rowBase + r) * HEADS + h] = sd;
    }
}

// ---------------------------------------------------------------------------
// Kernel 4: alpha[e,h] = leaky_relu(a_src[src]+a_dst[dst]+ef[e]·v[h]); atomic seg-max
// ---------------------------------------------------------------------------
__global__ void alpha_kernel(const float* __restrict__ ef,
                             const int* __restrict__ srcI,
                             const int* __restrict__ dstI,
                             const float* __restrict__ v_edge,
                             const float* __restrict__ a_src,
                             const float* __restrict__ a_dst,
                             float* __restrict__ alpha,
                             unsigned* __restrict__ nmax,
                             int E_) {
    int e = blockIdx.x * blockDim.x + threadIdx.x;
    if (e >= E_) return;
    const int s = srcI[e], d = dstI[e];
    float acc0 = 0.f, acc1 = 0.f, acc2 = 0.f, acc3 = 0.f;
    const float4* efr = (const float4*)(ef + (size_t)e * EDGE_F_);
    #pragma unroll
    for (int q = 0; q < EDGE_F_ / 4; ++q) {
        const float4 ev = efr[q];
        const float* v0 = v_edge + 4 * q;
        acc0 += ev.x * v0[0]      + ev.y * v0[1]      + ev.z * v0[2]      + ev.w * v0[3];
        acc1 += ev.x * v0[32]     + ev.y * v0[33]     + ev.z * v0[34]     + ev.w * v0[35];
        acc2 += ev.x * v0[64]     + ev.y * v0[65]     + ev.z * v0[66]     + ev.w * v0[67];
        acc3 += ev.x * v0[96]     + ev.y * v0[97]     + ev.z * v0[98]     + ev.w * v0[99];
    }
    float ae[4] = {acc0, acc1, acc2, acc3};
    #pragma unroll
    for (int h = 0; h < HEADS; ++h) {
        float a = ae[h] + a_src[(size_t)s * HEADS + h] + a_dst[(size_t)d * HEADS + h];
        a = (a > 0.f) ? a : NEG_SLOPE * a;
        alpha[(size_t)e * HEADS + h] = a;
        unsigned u = __float_as_uint(a);
        u = (u & 0x80000000u) ? ~u : (u | 0x80000000u);
        atomicMax(&nmax[(size_t)d * HEADS + h], u);
    }
}

// ---------------------------------------------------------------------------
// Kernel 5: ex = exp(alpha - m[dst]); denom[dst] += ex  (ex overwrites alpha)
// ---------------------------------------------------------------------------
__global__ void denom_kernel(const int* __restrict__ dstI,
                             float* __restrict__ alpha,
                             const unsigned* __restrict__ nmax,
                             float* __restrict__ denom,
                             int E_) {
    int e = blockIdx.x * blockDim.x + threadIdx.x;
    if (e >= E_) return;
    const int d = dstI[e];
    float4 a4 = ((const float4*)alpha)[e];
    float av[4] = {a4.x, a4.y, a4.z, a4.w};
    float ex[4];
    #pragma unroll
    for (int h = 0; h < HEADS; ++h) {
        unsigned u = nmax[(size_t)d * HEADS + h];
        float m = (u & 0x80000000u) ? __uint_as_float(u & 0x7FFFFFFFu)
                                    : __uint_as_float(~u);
        if (m < -3.0e38f) m = 0.f;       // reference: non-finite max -> 0
        ex[h] = expf(av[h] - m);
        atomicAdd(&denom[(size_t)d * HEADS + h], ex[h]);
    }
    ((float4*)alpha)[e] = make_float4(ex[0], ex[1], ex[2], ex[3]);
}

// ---------------------------------------------------------------------------
// Kernel 6: message scatter. One wave per edge, lane owns 4 channels:
//   out[dst, lane*4..+3] += attn(h) * x[src, lane*4..+3],  h = lane/8
// ---------------------------------------------------------------------------
__global__ void scatter_kernel(const int* __restrict__ srcI,
                               const int* __restrict__ dstI,
                               const float* __restrict__ ex,
                               const float* __restrict__ denom,
                               const float* __restrict__ x,
                               float* __restrict__ outv,
                               int E_) {
    long long gid = (long long)blockIdx.x * blockDim.x + threadIdx.x;
    int e = (int)(gid >> 5);
    if (e >= E_) return;
    const int lane = (int)(gid & 31);
    const int s = srcI[e], d = dstI[e];
    const int h = lane >> 3;
    const float attn = ex[(size_t)e * HEADS + h] /
                       (denom[(size_t)d * HEADS + h] + EPS_);
    const float4 xv = ((const float4*)(x + (size_t)s * HC))[lane];
    float* op = outv + (size_t)d * HC + lane * 4;
    atomicAdd(op + 0, attn * xv.x);
    atomicAdd(op + 1, attn * xv.y);
    atomicAdd(op + 2, attn * xv.z);
    atomicAdd(op + 3, attn * xv.w);
}

// ---------------------------------------------------------------------------
// Kernel 7: in-place ELU
// ---------------------------------------------------------------------------
__global__ void elu_kernel(float* __restrict__ o, long long n) {
    long long i = (long long)blockIdx.x * blockDim.x + threadIdx.x;
    if (i < n) {
        const float v = o[i];
        o[i] = v > 0.f ? v : expm1f(v);
    }
}

// ---------------------------------------------------------------------------
extern "C" void kernel_launch(void* const* d_in, const int* in_sizes, int n_in,
                              void* d_out, int out_size, void* d_ws, size_t ws_size,
                              hipStream_t stream) {
    const float* nf     = (const float*)d_in[0];   // [N,128]
    const float* ef     = (const float*)d_in[1];   // [E,32]
    const int*   ei     = (const int*)  d_in[2];   // [2,E]
    const float* Wg     = (const float*)d_in[3];   // [128,128]
    const float* attS   = (const float*)d_in[4];   // [4,32]
    const float* attD   = (const float*)d_in[5];   // [4,32]
    const float* attE   = (const float*)d_in[6];   // [4,32]
    const float* bias   = (const float*)d_in[7];   // [128]
    const float* We     = (const float*)d_in[8];   // [128,32]
    const float* Ws     = (const float*)d_in[9];   // [128,128]

    const int N = in_sizes[0] / IN_F;              // 50000 (divisible by 16)
    const int E = in_sizes[1] / EDGE_F_;           // 800000
    const int* srcI = ei;
    const int* dstI = ei + E;
    float* outv = (float*)d_out;

    // workspace carve-up (256B aligned slices)
    char* wsb = (char*)d_ws;
    size_t off = 0;
    auto carve = [&](size_t bytes) -> void* {
        void* p = wsb + off;
        off += (bytes + 255) & ~(size_t)255;
        return p;
    };
    float*    x      = (float*)   carve((size_t)N * HC * sizeof(float));     // 25.6 MB
    float*    alpha  = (float*)   carve((size_t)E * HEADS * sizeof(float));  // 12.8 MB
    float*    a_src  = (float*)   carve((size_t)N * HEADS * sizeof(float));
    float*    a_dst  = (float*)   carve((size_t)N * HEADS * sizeof(float));
    unsigned* nmax   = (unsigned*)carve((size_t)N * HEADS * sizeof(unsigned));
    float*    denom  = (float*)   carve((size_t)N * HEADS * sizeof(float));
    float*    v_edge = (float*)   carve((size_t)HEADS * EDGE_F_ * sizeof(float));
    (void)ws_size; (void)n_in; (void)out_size;

    fold_edge_kernel<<<1, 128, 0, stream>>>(We, attE, v_edge);
    init_kernel<<<(N * HEADS + 255) / 256, 256, 0, stream>>>(nmax, denom, N * HEADS);
    node_gemm_kernel<<<N / 16, 128, 0, stream>>>(nf, Wg, Ws, attS, attD, bias,
                                                 x, a_src, a_dst, outv);
    alpha_kernel<<<(E + 255) / 256, 256, 0, stream>>>(ef, srcI, dstI, v_edge,
                                                      a_src, a_dst, alpha, nmax, E);
    denom_kernel<<<(E + 255) / 256, 256, 0, stream>>>(dstI, alpha, nmax, denom, E);
    {
        long long tot = (long long)E * 32;
        scatter_kernel<<<(unsigned)((tot + 255) / 256), 256, 0, stream>>>(
            srcI, dstI, alpha, denom, x, outv, E);
    }
    {
        long long tot = (long long)N * HC;
        elu_kernel<<<(unsigned)((tot + 255) / 256), 256, 0, stream>>>(outv, tot);
    }
}